// LabelGuidedPrototypeLoss_19241453486093
// MI455X (gfx1250) — compile-verified
//
#include <hip/hip_runtime.h>

typedef __attribute__((ext_vector_type(2))) float v2f;
typedef __attribute__((ext_vector_type(8))) float v8f;

// Problem constants (fixed by reference setup_inputs)
constexpr int B = 2;
constexpr int M = 4;
constexpr int C = 64;
constexpr int K = 4;              // NUM_CLASSES
constexpr int V = 48 * 48 * 48;   // 110592
constexpr int V4 = V / 4;         // 27648 float4s
constexpr float EPS       = 1e-6f;
constexpr float NORM_EPS  = 1e-12f;
constexpr float DELTA_POS = 0.9f;
constexpr float DELTA_NEG = 0.1f;
// IGNORE_INDEX = 0, MIN_PIXELS = 1, LAMBDA_* = 1

// Phase-1 tiling
constexpr int TPB    = 256;
constexpr int CG     = 8;                 // c-values per block (feats read once; w/tgt re-read 8x)
constexpr int VIT4   = 4;                 // float4 iterations per thread
constexpr int NCHUNK = V4 / (TPB * VIT4); // 27 (exact: 27*1024 = 27648)

// workspace layout (floats)
//   [0,    2048)  num[B][M][K][C]
//   [2048, 2080)  tw [B][M][K]   (includes +EPS)
//   [2080, 2088)  cnt[B][K]
//   [2088, 2092)  scal: {alignSum, alignCnt, sepSum, sepCnt}

// ---------------------------------------------------------------------------
// Kernel 1: tw[b,m,k] = sum_v w*(tgt==k) + EPS ;  cnt[b,k] = sum_v (tgt==k)
// one block per (b,m); b128 vectorized stream
// ---------------------------------------------------------------------------
__global__ void lgpl_twcount(const float* __restrict__ w,
                             const int*   __restrict__ tgt,
                             float* __restrict__ tw,
                             float* __restrict__ cnt) {
  const int bm = blockIdx.x;          // b*M + m
  const int b  = bm / M;
  const bool doCnt = (bm % M) == 0;   // count is independent of m
  const float4* wp4 = (const float4*)(w   + (size_t)bm * V);
  const int4*   tp4 = (const int4*)  (tgt + (size_t)b  * V);

  float lw[K] = {0.f, 0.f, 0.f, 0.f};
  float lc[K] = {0.f, 0.f, 0.f, 0.f};
  for (int q = threadIdx.x; q < V4; q += blockDim.x) {
    const int4   t4 = tp4[q];
    const float4 w4 = wp4[q];
#pragma unroll
    for (int k = 0; k < K; ++k) {
      lw[k] += ((t4.x == k) ? w4.x : 0.f) + ((t4.y == k) ? w4.y : 0.f) +
               ((t4.z == k) ? w4.z : 0.f) + ((t4.w == k) ? w4.w : 0.f);
      lc[k] += ((t4.x == k) ? 1.f : 0.f) + ((t4.y == k) ? 1.f : 0.f) +
               ((t4.z == k) ? 1.f : 0.f) + ((t4.w == k) ? 1.f : 0.f);
    }
  }

  __shared__ float s[2 * K];
  if (threadIdx.x < 2 * K) s[threadIdx.x] = 0.f;
  __syncthreads();
#pragma unroll
  for (int k = 0; k < K; ++k) {
    atomicAdd(&s[k], lw[k]);
    if (doCnt) atomicAdd(&s[K + k], lc[k]);
  }
  __syncthreads();
  if (threadIdx.x < K) {
    tw[bm * K + threadIdx.x] = s[threadIdx.x] + EPS;
    if (doCnt) cnt[b * K + threadIdx.x] = s[K + threadIdx.x];
  }
}

// ---------------------------------------------------------------------------
// Kernel 2: num[b,m,k,c] += sum_v f[b,m,c,v]*w[b,m,v]*(tgt[b,v]==k)
// grid = (NCHUNK, C/CG, B*M). feats streamed once via global_load_b128:
// each lane loads a float4, a wave covers 512 contiguous bytes per load.
// ---------------------------------------------------------------------------
__global__ void lgpl_protosum(const float* __restrict__ f,
                              const float* __restrict__ w,
                              const int*   __restrict__ tgt,
                              float* __restrict__ num) {
  const int bm = blockIdx.z;
  const int b  = bm >> 2;                 // M == 4
  const int c0 = blockIdx.y * CG;
  const int qbase = blockIdx.x * (TPB * VIT4) + threadIdx.x;

  const float4* wp4 = (const float4*)(w   + (size_t)bm * V);
  const int4*   tp4 = (const int4*)  (tgt + (size_t)b  * V);
  const float4* fp4 = (const float4*)(f + ((size_t)bm * C + c0) * V);

  float acc[CG][K];
#pragma unroll
  for (int cc = 0; cc < CG; ++cc)
#pragma unroll
    for (int k = 0; k < K; ++k) acc[cc][k] = 0.f;

  for (int it = 0; it < VIT4; ++it) {
    const int q = qbase + it * TPB;
    const int4   t4 = tp4[q];
    const float4 w4 = wp4[q];
    // per-element, per-class masked weights
    float sx[K], sy[K], sz[K], sw[K];
#pragma unroll
    for (int k = 0; k < K; ++k) {
      sx[k] = (t4.x == k) ? w4.x : 0.f;
      sy[k] = (t4.y == k) ? w4.y : 0.f;
      sz[k] = (t4.z == k) ? w4.z : 0.f;
      sw[k] = (t4.w == k) ? w4.w : 0.f;
    }
#pragma unroll
    for (int cc = 0; cc < CG; ++cc) {
      const float4 f4 = fp4[(size_t)cc * V4 + q];
#pragma unroll
      for (int k = 0; k < K; ++k) {
        float a = acc[cc][k];
        a = fmaf(f4.x, sx[k], a);
        a = fmaf(f4.y, sy[k], a);
        a = fmaf(f4.z, sz[k], a);
        a = fmaf(f4.w, sw[k], a);
        acc[cc][k] = a;
      }
    }
  }

  __shared__ float s[CG * K];
  if (threadIdx.x < CG * K) s[threadIdx.x] = 0.f;
  __syncthreads();
#pragma unroll
  for (int cc = 0; cc < CG; ++cc)
#pragma unroll
    for (int k = 0; k < K; ++k)
      atomicAdd(&s[cc * K + k], acc[cc][k]);
  __syncthreads();
  if (threadIdx.x < CG * K) {
    const int cc = threadIdx.x >> 2;
    const int k  = threadIdx.x & 3;
    atomicAdd(&num[(size_t)(bm * K + k) * C + c0 + cc], s[threadIdx.x]);
  }
}

// ---------------------------------------------------------------------------
// Kernel 3: per-batch finalize. One wave (32 lanes) per batch.
// Builds normalized prototypes P[16][64], computes Gram G = P * P^T with
// chained V_WMMA_F32_16X16X4_F32 (16 steps over C=64, full fp32), then the
// masked align/sep sums directly from the accumulator layout.
// ---------------------------------------------------------------------------
__global__ void lgpl_finalize(const float* __restrict__ num,
                              const float* __restrict__ tw,
                              const float* __restrict__ cnt,
                              float* __restrict__ scal) {
  __shared__ float P[16][68];   // pad 68: bank-conflict-free column reads
  __shared__ float om[16];
  __shared__ int   vld[16];

  const int lane = threadIdx.x;
  const int b    = blockIdx.x;

  if (lane < 16) {
    const int m = lane >> 2;
    const int k = lane & 3;
    const float t = tw[(b * M + m) * K + k];     // already +EPS
    const float c = cnt[b * K + k];
    om[lane]  = t / (c + EPS);
    vld[lane] = (c >= 1.0f) && (k != 0);         // MIN_PIXELS, IGNORE_INDEX
    const float* np = num + (size_t)((b * M + m) * K + k) * C;
    float ss = 0.f;
    const float inv_t = 1.0f / t;
    for (int cc = 0; cc < C; ++cc) {
      const float p = np[cc] * inv_t;
      P[lane][cc] = p;
      ss = fmaf(p, p, ss);
    }
    float nrm = sqrtf(fmaxf(ss, NORM_EPS * NORM_EPS));
    nrm = fmaxf(nrm, NORM_EPS);
    const float inv = 1.0f / nrm;
    for (int cc = 0; cc < C; ++cc) P[lane][cc] *= inv;
  }
  __syncthreads();

  // Gram via WMMA f32 16x16x4, K-chained over C=64.
  // A (16x4 f32): lanes 0-15 hold M=row, VGPR0=K0/VGPR1=K1; lanes 16-31 K2/K3.
  // B (4x16 f32) is the mirror layout; for a symmetric Gram both operands are
  // the identical per-lane register pair.
  const int row = lane & 15;
  const int kk  = (lane >> 4) << 1;
  v8f g = {};
#pragma unroll
  for (int ch = 0; ch < 16; ++ch) {
    const int c0 = ch * 4;
    v2f a;
    a.x = P[row][c0 + kk];
    a.y = P[row][c0 + kk + 1];
    g = __builtin_amdgcn_wmma_f32_16x16x4_f32(false, a, false, a,
                                              (short)0, g, false, false);
  }

  // D layout: VGPR r -> lanes 0-15: G[r][lane]; lanes 16-31: G[r+8][lane-16]
  const int ncol  = lane & 15;
  const int mbase = (lane >> 4) * 8;
  float aS = 0.f, aC = 0.f, sS = 0.f, sC = 0.f;
#pragma unroll
  for (int r = 0; r < 8; ++r) {
    const int i = mbase + r;     // row (m,k) index: m=i>>2, k=i&3
    const int j = ncol;
    const float gij = g[r];
    const int mi = i >> 2, ki = i & 3;
    const int mj = j >> 2, kj = j & 3;
    if (vld[i] && vld[j]) {
      if (ki == kj && mi < mj) {       // align: same class, modality pair m<n
        aC += 1.0f;
        aS += om[i] * om[j] * fmaxf(DELTA_POS - gij, 0.0f);
      }
      if (mi == mj && ki < kj) {       // sep: same modality, class pair k<l
        sC += 1.0f;
        sS += fmaxf(gij - DELTA_NEG, 0.0f);
      }
    }
  }

  // wave32 reduction
  for (int off = 16; off > 0; off >>= 1) {
    aS += __shfl_down(aS, off);
    aC += __shfl_down(aC, off);
    sS += __shfl_down(sS, off);
    sC += __shfl_down(sC, off);
  }
  if (lane == 0) {
    atomicAdd(&scal[0], aS);
    atomicAdd(&scal[1], aC);
    atomicAdd(&scal[2], sS);
    atomicAdd(&scal[3], sC);
  }
}

// ---------------------------------------------------------------------------
// Kernel 4: combine to scalar loss
// ---------------------------------------------------------------------------
__global__ void lgpl_combine(const float* __restrict__ scal,
                             float* __restrict__ out) {
  out[0] = scal[0] / (scal[1] + EPS) + scal[2] / (scal[3] + EPS);
}

// ---------------------------------------------------------------------------
extern "C" void kernel_launch(void* const* d_in, const int* in_sizes, int n_in,
                              void* d_out, int out_size, void* d_ws, size_t ws_size,
                              hipStream_t stream) {
  const float* feats   = (const float*)d_in[0];   // [B,M,C,D,H,W]
  const float* weights = (const float*)d_in[1];   // [B,M,D,H,W]
  const int*   target  = (const int*)  d_in[2];   // [B,1,D,H,W]
  float* out = (float*)d_out;

  float* ws   = (float*)d_ws;
  float* num  = ws;                 // 2048
  float* tw   = ws + 2048;          // 32
  float* cnt  = ws + 2080;          // 8
  float* scal = ws + 2088;          // 4

  hipMemsetAsync(d_ws, 0, 2092 * sizeof(float), stream);

  lgpl_twcount<<<B * M, TPB, 0, stream>>>(weights, target, tw, cnt);

  lgpl_protosum<<<dim3(NCHUNK, C / CG, B * M), TPB, 0, stream>>>(
      feats, weights, target, num);

  lgpl_finalize<<<B, 32, 0, stream>>>(num, tw, cnt, scal);

  lgpl_combine<<<1, 1, 0, stream>>>(scal, out);

  (void)in_sizes; (void)n_in; (void)out_size; (void)ws_size; (void)out;
}